// GCN_84000970375232
// MI455X (gfx1250) — compile-verified
//
#include <hip/hip_runtime.h>
#include <hip/hip_bf16.h>
#include <stdint.h>

typedef __attribute__((ext_vector_type(2))) float v2f;
typedef __attribute__((ext_vector_type(8))) float v8f;

#define N_NODES  50000
#define N_EDGES  800000
#define HID      128
#define N_LAYERS 3
#define N_CLASSES 40
#define LN_EPS   1e-5f

// ---- CDNA5 async global->LDS helpers (ASYNCcnt-tracked, 16B per lane) ----
__device__ __forceinline__ unsigned lds_off(const void* p) {
    // generic LDS address = {shared aperture, lds_offset}; low 32 bits = LDS offset
    return (unsigned)(uintptr_t)p;
}
__device__ __forceinline__ void async_copy16(unsigned lds_addr, const void* gptr) {
    asm volatile("global_load_async_to_lds_b128 %0, %1, off"
                 :: "v"(lds_addr), "v"((unsigned long long)(uintptr_t)gptr)
                 : "memory");
}
__device__ __forceinline__ void wait_async0() {
    asm volatile("s_wait_asynccnt 0" ::: "memory");
}

// ---------------- degree / normalization ----------------
__global__ void init_deg_kernel(float* __restrict__ deg) {
    int i = blockIdx.x * blockDim.x + threadIdx.x;
    if (i < N_NODES) deg[i] = 1.0f;                 // self-loop
}

__global__ void accum_deg_kernel(const long long* __restrict__ dst,
                                 float* __restrict__ deg) {
    int e = blockIdx.x * blockDim.x + threadIdx.x;
    if (e < N_EDGES) atomicAdd(&deg[(int)dst[e]], 1.0f);
}

__global__ void finalize_dinv_kernel(float* __restrict__ deg) {
    int i = blockIdx.x * blockDim.x + threadIdx.x;
    if (i < N_NODES) deg[i] = rsqrtf(deg[i]);
}

// ---------------- WMMA GEMM: H[N,128] = X[N,128] @ W[128,128] ----------------
// One block = one 16-row tile of X; 8 waves cover the 8 column tiles.
// W staged in double-buffered 32x128 LDS chunks via async b128 copies so the
// next chunk's fill overlaps the current chunk's 8 WMMAs.
// f32 WMMA 16x16x4 layouts: A: M=lane%16, K=2*(lane/16)+vgpr; B: N=lane%16,
// K=2*(lane/16)+vgpr; C: M=vgpr+8*(lane/16), N=lane%16.
__global__ void __launch_bounds__(256)
gemm128_wmma_kernel(const float* __restrict__ X, const float* __restrict__ W,
                    float* __restrict__ H) {
    __shared__ float sX[16 * 128];        // 8 KB
    __shared__ float sW[2][32 * 128];     // 2 x 16 KB double buffer

    const int tid  = threadIdx.x;
    const int wave = tid >> 5;            // 0..7 -> column tile
    const int lane = tid & 31;
    const int m0   = blockIdx.x * 16;

    const unsigned sx_l  = lds_off(sX);
    const unsigned sw_l0 = lds_off(sW[0]);
    const unsigned sw_l1 = lds_off(sW[1]);

    // prologue: async-fill X tile (512 x 16B) and W chunk 0 (1024 x 16B)
    #pragma unroll
    for (int i = tid; i < 512; i += 256)
        async_copy16(sx_l + i * 16, X + (size_t)m0 * 128 + i * 4);
    #pragma unroll
    for (int i = tid; i < 1024; i += 256)
        async_copy16(sw_l0 + i * 16, W + i * 4);
    wait_async0();
    __syncthreads();

    const int row  = lane & 15;
    const int half = lane >> 4;
    const int col  = wave * 16 + (lane & 15);

    v8f acc = {};
    for (int kb = 0; kb < 4; ++kb) {
        const int cur = kb & 1;
        if (kb < 3) {                     // prefetch next W chunk into other buffer
            const unsigned dst = cur ? sw_l0 : sw_l1;
            #pragma unroll
            for (int i = tid; i < 1024; i += 256)
                async_copy16(dst + i * 16, W + (size_t)(kb + 1) * 4096 + i * 4);
        }
        const float* w = sW[cur];
        #pragma unroll
        for (int kk = 0; kk < 32; kk += 4) {
            const int K0 = kk + 2 * half;
            v2f a = { sX[row * 128 + kb * 32 + K0],
                      sX[row * 128 + kb * 32 + K0 + 1] };
            v2f b = { w[K0 * 128 + col],
                      w[(K0 + 1) * 128 + col] };
            acc = __builtin_amdgcn_wmma_f32_16x16x4_f32(
                      false, a, false, b, (short)0, acc, false, false);
        }
        wait_async0();                    // this wave's prefetch landed
        __syncthreads();                  // everyone's prefetch landed / reads done
    }
    #pragma unroll
    for (int g = 0; g < 8; ++g) {
        const int r = g + 8 * half;
        H[(size_t)(m0 + r) * 128 + col] = acc[g];
    }
}

// ---------------- agg init: agg = dinv^2 * h + bias ----------------
__global__ void init_agg_kernel(const float* __restrict__ h,
                                const float* __restrict__ dinv,
                                const float* __restrict__ bias,
                                float* __restrict__ agg) {
    int i = blockIdx.x * blockDim.x + threadIdx.x;
    if (i < N_NODES * HID) {
        int n = i >> 7, f = i & 127;
        float di = dinv[n];
        agg[i] = di * di * h[i] + bias[f];
    }
}

// ---------------- edge scatter: agg[dst] += norm * h[src] ----------------
__global__ void __launch_bounds__(256)
scatter_edges_kernel(const long long* __restrict__ ei,
                     const float* __restrict__ h,
                     const float* __restrict__ dinv,
                     float* __restrict__ agg) {
    int gw   = (blockIdx.x * blockDim.x + threadIdx.x) >> 5;
    int lane = threadIdx.x & 31;
    if (gw >= N_EDGES) return;
    int s = (int)ei[gw];
    int d = (int)ei[N_EDGES + gw];
    float norm = dinv[s] * dinv[d];
    float4 v = ((const float4*)(h + (size_t)s * HID))[lane];
    float* ad = agg + (size_t)d * HID + lane * 4;
    atomicAdd(ad + 0, norm * v.x);
    atomicAdd(ad + 1, norm * v.y);
    atomicAdd(ad + 2, norm * v.z);
    atomicAdd(ad + 3, norm * v.w);
}

// ---------------- LayerNorm + ReLU (wave per node, wave32 shuffles) ----------------
__global__ void __launch_bounds__(256)
ln_relu_kernel(const float* __restrict__ agg, const float* __restrict__ g,
               const float* __restrict__ b, float* __restrict__ out) {
    int node = (blockIdx.x * blockDim.x + threadIdx.x) >> 5;
    int lane = threadIdx.x & 31;
    if (node >= N_NODES) return;
    float4 v = ((const float4*)(agg + (size_t)node * HID))[lane];
    float s = v.x + v.y + v.z + v.w;
    #pragma unroll
    for (int m = 16; m; m >>= 1) s += __shfl_xor(s, m, 32);
    float mu = s * (1.0f / 128.0f);
    float dx = v.x - mu, dy = v.y - mu, dz = v.z - mu, dw = v.w - mu;
    float q = dx * dx + dy * dy + dz * dz + dw * dw;
    #pragma unroll
    for (int m = 16; m; m >>= 1) q += __shfl_xor(q, m, 32);
    float rstd = rsqrtf(q * (1.0f / 128.0f) + LN_EPS);
    int f = lane * 4;
    float4 gg = *(const float4*)(g + f);
    float4 bb = *(const float4*)(b + f);
    float4 o;
    o.x = fmaxf(dx * rstd * gg.x + bb.x, 0.0f);
    o.y = fmaxf(dy * rstd * gg.y + bb.y, 0.0f);
    o.z = fmaxf(dz * rstd * gg.z + bb.z, 0.0f);
    o.w = fmaxf(dw * rstd * gg.w + bb.w, 0.0f);
    ((float4*)(out + (size_t)node * HID))[lane] = o;
}

// ---------------- classifier: out[N,40] = X[N,128] @ Wout[128,40] + bout ----------------
// Wout staged unpadded (128x40 = 1280 x 16B, contiguous). Columns >= 40 use a
// clamped LDS index: B column n only contributes to D column n, never stored.
__global__ void __launch_bounds__(128)
classify_wmma_kernel(const float* __restrict__ X, const float* __restrict__ Wo,
                     const float* __restrict__ bo, float* __restrict__ out) {
    __shared__ float sX[16 * 128];    // 8 KB
    __shared__ float sW[128 * 40];    // 20 KB
    const int tid  = threadIdx.x;
    const int wave = tid >> 5;
    const int lane = tid & 31;
    const int m0   = blockIdx.x * 16;

    const unsigned sx_l = lds_off(sX);
    const unsigned sw_l = lds_off(sW);
    #pragma unroll
    for (int i = tid; i < 512; i += 128)
        async_copy16(sx_l + i * 16, X + (size_t)m0 * 128 + i * 4);
    #pragma unroll
    for (int i = tid; i < 1280; i += 128)
        async_copy16(sw_l + i * 16, Wo + i * 4);
    wait_async0();
    __syncthreads();

    if (wave < 3) {                   // 3 column tiles cover 48 (>=40) cols
        const int row  = lane & 15;
        const int half = lane >> 4;
        const int col  = wave * 16 + (lane & 15);
        const int cc   = (col < N_CLASSES) ? col : 0;   // clamp for pad lanes
        v8f acc = {};
        #pragma unroll
        for (int kk = 0; kk < 128; kk += 4) {
            const int K0 = kk + 2 * half;
            v2f a = { sX[row * 128 + K0], sX[row * 128 + K0 + 1] };
            v2f b = { sW[K0 * 40 + cc], sW[(K0 + 1) * 40 + cc] };
            acc = __builtin_amdgcn_wmma_f32_16x16x4_f32(
                      false, a, false, b, (short)0, acc, false, false);
        }
        if (col < N_CLASSES) {
            float bias = bo[col];
            #pragma unroll
            for (int g = 0; g < 8; ++g) {
                const int r = g + 8 * half;
                out[(size_t)(m0 + r) * N_CLASSES + col] = acc[g] + bias;
            }
        }
    }
}

// ---------------- launch ----------------
extern "C" void kernel_launch(void* const* d_in, const int* in_sizes, int n_in,
                              void* d_out, int out_size, void* d_ws, size_t ws_size,
                              hipStream_t stream) {
    (void)in_sizes; (void)n_in; (void)out_size; (void)ws_size;
    const float*     x      = (const float*)d_in[0];
    const long long* ei     = (const long long*)d_in[1];   // int64 [2, E]
    const float*     conv_w = (const float*)d_in[2];       // [3,128,128]
    const float*     conv_b = (const float*)d_in[3];       // [3,128]
    const float*     ln_g   = (const float*)d_in[4];
    const float*     ln_b   = (const float*)d_in[5];
    const float*     wout   = (const float*)d_in[6];       // [128,40]
    const float*     bout   = (const float*)d_in[7];       // [40]
    float*           out    = (float*)d_out;

    char* ws = (char*)d_ws;
    size_t off = 0;
    auto take = [&](size_t bytes) -> float* {
        float* p = (float*)(ws + off);
        off = (off + bytes + 255) & ~(size_t)255;
        return p;
    };
    float* dinv = take((size_t)N_NODES * sizeof(float));
    float* h    = take((size_t)N_NODES * HID * sizeof(float));
    float* agg  = take((size_t)N_NODES * HID * sizeof(float));
    float* buf  = take((size_t)N_NODES * HID * sizeof(float));

    // degrees -> dinv
    init_deg_kernel<<<(N_NODES + 255) / 256, 256, 0, stream>>>(dinv);
    accum_deg_kernel<<<(N_EDGES + 255) / 256, 256, 0, stream>>>(ei + N_EDGES, dinv);
    finalize_dinv_kernel<<<(N_NODES + 255) / 256, 256, 0, stream>>>(dinv);

    const float* xin = x;
    for (int l = 0; l < N_LAYERS; ++l) {
        gemm128_wmma_kernel<<<N_NODES / 16, 256, 0, stream>>>(
            xin, conv_w + (size_t)l * HID * HID, h);
        init_agg_kernel<<<(N_NODES * HID + 255) / 256, 256, 0, stream>>>(
            h, dinv, conv_b + (size_t)l * HID, agg);
        scatter_edges_kernel<<<((size_t)N_EDGES * 32 + 255) / 256, 256, 0, stream>>>(
            ei, h, dinv, agg);
        ln_relu_kernel<<<((size_t)N_NODES * 32 + 255) / 256, 256, 0, stream>>>(
            agg, ln_g + (size_t)l * HID, ln_b + (size_t)l * HID, buf);
        xin = buf;
    }
    classify_wmma_kernel<<<N_NODES / 16, 128, 0, stream>>>(buf, wout, bout, out);
}